// NeuralNetwork_89833535963626
// MI455X (gfx1250) — compile-verified
//
#include <hip/hip_runtime.h>
#include <math.h>

// float32(np.pi) and float32(2*np.pi) — decimal literals round to the same
// binary32 values the reference uses.
#define PI_F          3.14159265358979323846f   /* 0x40490FDB */
#define TWO_PI_F      6.28318530717958647692f   /* 0x40C90FDB */
#define INV_TWO_PI_F  0.15915494309189533577f   /* 0x3E22F983 */

#define HIDDEN 10

// gfx1250 has V_TANH_F32 (single-instruction transcendental tanh).
#if __has_builtin(__builtin_amdgcn_tanhf)
__device__ __forceinline__ float fast_tanh(float v) { return __builtin_amdgcn_tanhf(v); }
#else
__device__ __forceinline__ float fast_tanh(float v) { return tanhf(v); }
#endif

// Wrap x to [-pi, pi), matching jnp.mod(x + pi, 2pi) - pi semantics
// (mod result in [0, 2pi), i.e. sign of divisor).
__device__ __forceinline__ float wrap_angle(float x) {
  float r = x + PI_F;
  float q = truncf(r * INV_TWO_PI_F);
  float m = fmaf(-q, TWO_PI_F, r);          // single-rounded r - q*2pi
  m = (m < 0.0f)      ? m + TWO_PI_F : m;   // fix q overshoot
  m = (m >= TWO_PI_F) ? m - TWO_PI_F : m;   // fix q undershoot
  return m - PI_F;
}

typedef float f4 __attribute__((ext_vector_type(4)));

__device__ __forceinline__ float mlp_one(float xin,
                                         const float* W1, const float* B1,
                                         const float* W2, float B2) {
  float a = wrap_angle(xin);
  float acc = B2;
#pragma unroll
  for (int j = 0; j < HIDDEN; ++j) {
    acc = fmaf(W2[j], fast_tanh(fmaf(a, W1[j], B1[j])), acc);
  }
  return acc;
}

// Main streaming kernel: one float4 chunk per thread, straight-line body.
__global__ __launch_bounds__(256)
void NeuralNetwork_89833535963626_kernel(const float* __restrict__ x,
                                         const float* __restrict__ w1,
                                         const float* __restrict__ b1,
                                         const float* __restrict__ w2,
                                         const float* __restrict__ b2,
                                         float* __restrict__ out,
                                         int nvec) {
  const int i = blockIdx.x * blockDim.x + threadIdx.x;
  if (i >= nvec) return;

  // Uniform-address weight loads -> s_load broadcasts into SGPRs.
  float W1[HIDDEN], B1[HIDDEN], W2[HIDDEN];
#pragma unroll
  for (int j = 0; j < HIDDEN; ++j) {
    W1[j] = w1[j];
    B1[j] = b1[j];
    W2[j] = w2[j];
  }
  const float B2 = b2[0];

  // 128-bit nontemporal streaming load/store: data is touched exactly once.
  const f4 xv = __builtin_nontemporal_load(reinterpret_cast<const f4*>(x) + i);
  f4 ov;
  ov.x = mlp_one(xv.x, W1, B1, W2, B2);
  ov.y = mlp_one(xv.y, W1, B1, W2, B2);
  ov.z = mlp_one(xv.z, W1, B1, W2, B2);
  ov.w = mlp_one(xv.w, W1, B1, W2, B2);
  __builtin_nontemporal_store(ov, reinterpret_cast<f4*>(out) + i);
}

// Tail kernel for n % 4 != 0 (never launched for the reference N = 4194304).
__global__ __launch_bounds__(32)
void NeuralNetwork_89833535963626_tail(const float* __restrict__ x,
                                       const float* __restrict__ w1,
                                       const float* __restrict__ b1,
                                       const float* __restrict__ w2,
                                       const float* __restrict__ b2,
                                       float* __restrict__ out,
                                       int start, int n) {
  const int k = start + (int)threadIdx.x;
  if (k >= n) return;
  float W1[HIDDEN], B1[HIDDEN], W2[HIDDEN];
#pragma unroll
  for (int j = 0; j < HIDDEN; ++j) {
    W1[j] = w1[j];
    B1[j] = b1[j];
    W2[j] = w2[j];
  }
  out[k] = mlp_one(x[k], W1, B1, W2, b2[0]);
}

extern "C" void kernel_launch(void* const* d_in, const int* in_sizes, int n_in,
                              void* d_out, int out_size, void* d_ws, size_t ws_size,
                              hipStream_t stream) {
  const float* x  = (const float*)d_in[0];
  const float* w1 = (const float*)d_in[1];
  const float* b1 = (const float*)d_in[2];
  const float* w2 = (const float*)d_in[3];
  const float* b2 = (const float*)d_in[4];
  float* out = (float*)d_out;

  const int n    = in_sizes[0];          // 4194304 elements
  const int nvec = n >> 2;               // full float4 chunks
  const int block = 256;                 // 8 wave32s

  if (nvec > 0) {
    const int grid = (nvec + block - 1) / block;   // exact: 4096 blocks
    NeuralNetwork_89833535963626_kernel<<<grid, block, 0, stream>>>(
        x, w1, b1, w2, b2, out, nvec);
  }
  if (n & 3) {
    NeuralNetwork_89833535963626_tail<<<1, 32, 0, stream>>>(
        x, w1, b1, w2, b2, out, nvec << 2, n);
  }
}